// SFAoatt_73143293051365
// MI455X (gfx1250) — compile-verified
//
#include <hip/hip_runtime.h>

// Choice MoE weighted-sum: out[b,c,hw] = sum_e l[e,b,c,hw] * g[b, e*C + c]
// Memory-bound (0.44 FLOP/byte) -> pinned to 23.3 TB/s HBM roof (~6.5us).
// CDNA5 paths: GLOBAL_LOAD_ASYNC_TO_LDS_B128 + S_WAIT_ASYNCCNT double-buffer,
// and the E-reduction computed on V_WMMA_F32_16X16X4_F32 (broadcast-B trick:
// B[k,n] = w[e_k] for all n, so every D column is the exact 16-hw output).
// Two chained WMMAs cover E=8; global bytes identical to the scalar version.

#define NE 8
#define NB 8
#define NC 512
#define NHW 1024               // H*W = 32*32
#define TPB 256                // 8 waves (wave32)
#define TILES_PER_BLK 8
#define NTILES (NB * NC)       // 4096 (b,c) tiles
#define NBLK (NTILES / TILES_PER_BLK)  // 512 workgroups

typedef __attribute__((ext_vector_type(2))) float v2f;
typedef __attribute__((ext_vector_type(8))) float v8f;

// ---- CDNA5 async global->LDS copy (gfx1250, tracked by ASYNCcnt) ----
__device__ __forceinline__ void async_load_b128_to_lds(unsigned lds_byte_off,
                                                       const float* gsrc) {
  asm volatile("global_load_async_to_lds_b128 %0, %1, off"
               :
               : "v"(lds_byte_off),
                 "v"((unsigned long long)(size_t)gsrc)
               : "memory");
}
__device__ __forceinline__ void wait_asynccnt_le8() {
  asm volatile("s_wait_asynccnt 0x8" ::: "memory");
}
__device__ __forceinline__ void wait_asynccnt_0() {
  asm volatile("s_wait_asynccnt 0x0" ::: "memory");
}

// Flat pointer -> LDS byte offset: flat-LDS aperture keeps offset in addr[31:0].
__device__ __forceinline__ unsigned lds_offset_of(const void* p) {
  return (unsigned)(size_t)p;
}

__global__ __launch_bounds__(TPB) void choice_moe_wmma_kernel(
    const float* __restrict__ l,   // [E, B, C, HW]
    const float* __restrict__ g,   // [B, E*C]
    float* __restrict__ out) {     // [B, C, HW]
  __shared__ __align__(16) float lds[2][NE * NHW];  // 2 x 32 KB

  const int t    = threadIdx.x;
  const int lane = t & 31;        // wave32
  const int wv   = t >> 5;        // wave id, 0..7
  const int half = lane >> 4;     // 0: lanes 0-15, 1: lanes 16-31
  const int n16  = lane & 15;
  const int tile0 = (int)blockIdx.x * TILES_PER_BLK;

  // Per-lane base dword index into the staged [e][hw] tile for WMMA A operands:
  // A-matrix f32 16x4 layout: lanes 0-15 carry K=0,1 (VGPR0,1); lanes 16-31 K=2,3.
  // K indexes experts, so the upper half-wave reads experts e+2 (=> +2048 dwords).
  const int abase = half * 2048 + n16;

  // ---- prologue: stage tile 0 into buffer 0 ----
  {
    const int j = tile0;
    const int b = j >> 9;          // NC == 512
    const int c = j & (NC - 1);
    const unsigned dbase = lds_offset_of(&lds[0][0]);
#pragma unroll
    for (int e = 0; e < NE; ++e) {
      const float* src = l + ((((size_t)e * NB + b) * NC + c) * NHW) + (size_t)t * 4;
      async_load_b128_to_lds(dbase + (unsigned)((e * NHW + t * 4) * sizeof(float)), src);
    }
  }

  for (int i = 0; i < TILES_PER_BLK; ++i) {
    const int j = tile0 + i;
    const int b = j >> 9;
    const int c = j & (NC - 1);

    // ---- stage next tile (double buffer), then wait for current tile ----
    if (i + 1 < TILES_PER_BLK) {
      const int j2 = j + 1;
      const int b2 = j2 >> 9;
      const int c2 = j2 & (NC - 1);
      const unsigned dbase = lds_offset_of(&lds[(i + 1) & 1][0]);
#pragma unroll
      for (int e = 0; e < NE; ++e) {
        const float* src = l + ((((size_t)e * NB + b2) * NC + c2) * NHW) + (size_t)t * 4;
        async_load_b128_to_lds(dbase + (unsigned)((e * NHW + t * 4) * sizeof(float)), src);
      }
      wait_asynccnt_le8();  // 8 outstanding = next tile only -> current tile done
    } else {
      wait_asynccnt_0();
    }
    __syncthreads();

    // ---- gate weights: uniform across workgroup -> scalar loads ----
    float w[NE];
#pragma unroll
    for (int e = 0; e < NE; ++e) {
      w[e] = g[(size_t)b * (NE * NC) + (size_t)e * NC + c];
    }

    // Broadcast-B operands (4x16 f32 layout: VGPR0 = rows K0|K1, VGPR1 = K2|K3):
    v2f b0, b1;
    b0.x = half ? w[1] : w[0];
    b0.y = half ? w[3] : w[2];
    b1.x = half ? w[5] : w[4];
    b1.y = half ? w[7] : w[6];

    const float* Lb = &lds[i & 1][0];
    float* outTile = out + ((size_t)b * NC + c) * NHW;

    // Each wave owns 8 chunks of 16 hw positions: hw0 = wv*128 + q*16.
#pragma unroll
    for (int q = 0; q < 8; ++q) {
      const int hw0 = wv * 128 + q * 16;
      const int di  = abase + hw0;

      v2f a0, a1;
      a0.x = Lb[di];            // e = 0 + 2*half
      a0.y = Lb[di + 1024];     // e = 1 + 2*half
      a1.x = Lb[di + 4096];     // e = 4 + 2*half
      a1.y = Lb[di + 5120];     // e = 5 + 2*half

      v8f acc = {};
      acc = __builtin_amdgcn_wmma_f32_16x16x4_f32(false, a0, false, b0,
                                                  (short)0, acc, false, false);
      acc = __builtin_amdgcn_wmma_f32_16x16x4_f32(false, a1, false, b1,
                                                  (short)0, acc, false, false);

      // All D columns are identical: lane0 holds m=0..7, lane16 holds m=8..15.
      if (n16 == 0) {
        float* ob = outTile + hw0 + half * 8;
        float4 s0 = make_float4(acc[0], acc[1], acc[2], acc[3]);
        float4 s1 = make_float4(acc[4], acc[5], acc[6], acc[7]);
        *(float4*)(ob)     = s0;
        *(float4*)(ob + 4) = s1;
      }
    }

    __syncthreads();  // protect LDS buffer reuse two tiles ahead
  }
}

extern "C" void kernel_launch(void* const* d_in, const int* in_sizes, int n_in,
                              void* d_out, int out_size, void* d_ws, size_t ws_size,
                              hipStream_t stream) {
  (void)in_sizes; (void)n_in; (void)out_size; (void)d_ws; (void)ws_size;
  const float* l = (const float*)d_in[0];
  const float* g = (const float*)d_in[1];
  float* out = (float*)d_out;
  hipLaunchKernelGGL(choice_moe_wmma_kernel, dim3(NBLK), dim3(TPB), 0, stream,
                     l, g, out);
}